// GINLayer_89996744720582
// MI455X (gfx1250) — compile-verified
//
#include <hip/hip_runtime.h>
#include <hip/hip_bf16.h>

typedef __attribute__((ext_vector_type(2))) float v2f;
typedef __attribute__((ext_vector_type(8))) float v8f;

#define DHID 128
#define BN_EPS 1e-5f
#define TILES_PER_BLOCK 4

// -------------------- out[i] = (1+eps) * in[i] --------------------
__global__ __launch_bounds__(256) void gin_scale(const float* __restrict__ in,
                                                 float* __restrict__ out,
                                                 const float* __restrict__ eps,
                                                 int total) {
    int i = blockIdx.x * 256 + threadIdx.x;
    if (i < total) out[i] = (1.0f + eps[0]) * in[i];
}

// -------------------- scatter-add: out[dst[e]] += h[src[e]] --------------------
// one wave per edge; each lane handles 4 consecutive floats of the 128-wide row
__global__ __launch_bounds__(256) void gin_scatter(const float* __restrict__ h,
                                                   const int* __restrict__ src,
                                                   const int* __restrict__ dst,
                                                   float* __restrict__ out,
                                                   int E) {
    int e = blockIdx.x * 8 + (threadIdx.x >> 5);
    if (e >= E) return;
    int lane = threadIdx.x & 31;
    int s = src[e];
    int d = dst[e];
    const float4 v = *(const float4*)(h + (size_t)s * DHID + lane * 4);
    float* o = out + (size_t)d * DHID + lane * 4;
    unsafeAtomicAdd(o + 0, v.x);
    unsafeAtomicAdd(o + 1, v.y);
    unsafeAtomicAdd(o + 2, v.z);
    unsafeAtomicAdd(o + 3, v.w);
}

// -------------------- zero the BN accumulator (sum[128], sumsq[128]) --------------------
__global__ void gin_zero(float* __restrict__ p, int n) {
    int i = blockIdx.x * blockDim.x + threadIdx.x;
    if (i < n) p[i] = 0.0f;
}

// -------------------- WMMA GEMM: out = A[nrows x 128] @ W[128 x 128] + bias --------------------
// block = 256 threads = 8 waves; wave w owns the 16-column slice [16w, 16w+16).
// The wave's B fragment for the FULL K=128 reduction (64 floats/lane) is loaded
// ONCE into registers, then the block sweeps TILES_PER_BLOCK row tiles; the
// steady-state inner loop is ds_load + v_wmma_f32_16x16x4_f32 only.
// If sums != nullptr, accumulates per-column sum / sum-of-squares of the
// PRE-activation output (BN input) via f32 atomics; relu applied when requested.
__global__ __launch_bounds__(256) void gin_gemm(const float* __restrict__ A,
                                                const float* __restrict__ W,
                                                const float* __restrict__ bias,
                                                float* __restrict__ out,
                                                float* __restrict__ sums,
                                                int relu, int nrows) {
    __shared__ float lds[16 * DHID];

    const int t    = threadIdx.x;
    const int lane = t & 31;
    const int wv   = t >> 5;                 // 0..7 -> column tile
    const int m    = lane & 15;              // A row within tile
    const int kh   = (lane >> 4) * 2;        // K sub-offset: 0 (lanes 0-15) or 2 (lanes 16-31)
    const int n    = wv * 16 + (lane & 15);  // output column

    // ---- register-resident B fragment: whole K=128 slab for this wave's columns ----
    v2f breg[32];
#pragma unroll
    for (int s = 0; s < 32; ++s) {
        const int k = s * 4 + kh;
        breg[s].x = W[k * DHID + n];
        breg[s].y = W[(k + 1) * DHID + n];
    }
    const float bv = bias[n];
    const int hi   = lane >> 4;

    // A-tile staging indices (cooperative 16x128 load, 8 floats per thread)
    const int lr = t >> 4;         // 0..15 row
    const int lc = (t & 15) * 8;   // 0..120 col
    const float* __restrict__ arow = lds + m * DHID;

    const int tile0 = blockIdx.x * TILES_PER_BLOCK;
    for (int tile = 0; tile < TILES_PER_BLOCK; ++tile) {
        const int rb = (tile0 + tile) * 16;      // uniform across block
        if (rb >= nrows) break;

        // fetch this tile's A slice into registers (overlaps previous tile's WMMAs)
        float4 v0 = {0, 0, 0, 0}, v1 = {0, 0, 0, 0};
        if (rb + lr < nrows) {
            const float4* g = (const float4*)(A + (size_t)(rb + lr) * DHID + lc);
            v0 = g[0];
            v1 = g[1];
        }
        // prefetch next tile's slice (global_prefetch_b8)
        if (rb + 16 + lr < nrows)
            __builtin_prefetch(A + (size_t)(rb + 16 + lr) * DHID + lc, 0, 1);

        __syncthreads();   // previous iteration's LDS reads complete
        {
            float4* sp = (float4*)(lds + lr * DHID + lc);
            sp[0] = v0;
            sp[1] = v1;
        }
        __syncthreads();   // tile visible to all waves

        v8f acc = {};
#pragma unroll
        for (int s = 0; s < 32; ++s) {
            v2f a;
            a.x = arow[s * 4 + kh];
            a.y = arow[s * 4 + kh + 1];
            acc = __builtin_amdgcn_wmma_f32_16x16x4_f32(
                /*neg_a=*/false, a, /*neg_b=*/false, breg[s],
                /*c_mod=*/(short)0, acc, /*reuse_a=*/false, /*reuse_b=*/false);
        }

        // epilogue: D layout -> VGPR r: lanes 0-15 = row rb+r, lanes 16-31 = row rb+r+8
        float s1 = 0.0f, s2 = 0.0f;
#pragma unroll
        for (int r = 0; r < 8; ++r) {
            int row = rb + r + 8 * hi;
            if (row < nrows) {
                float v = acc[r] + bv;
                float w = relu ? (v > 0.0f ? v : 0.0f) : v;
                out[(size_t)row * DHID + n] = w;
                s1 += v;        // BN statistics use pre-activation values
                s2 += v * v;
            }
        }
        if (sums) {
            s1 += __shfl_xor(s1, 16, 32);
            s2 += __shfl_xor(s2, 16, 32);
            if (lane < 16) {
                unsafeAtomicAdd(&sums[n], s1);
                unsafeAtomicAdd(&sums[DHID + n], s2);
            }
        }
    }
}

// -------------------- BN (training-mode batch stats) + ReLU, in place --------------------
__global__ __launch_bounds__(256) void gin_bn(float* __restrict__ h,
                                              const float* __restrict__ sums,
                                              const float* __restrict__ g,
                                              const float* __restrict__ b,
                                              int nrows) {
    int i = blockIdx.x * 256 + threadIdx.x;
    if (i >= nrows * DHID) return;
    int n = i & (DHID - 1);
    float inv  = 1.0f / (float)nrows;
    float mean = sums[n] * inv;
    float var  = sums[DHID + n] * inv - mean * mean;
    float y = (h[i] - mean) * rsqrtf(var + BN_EPS) * g[n] + b[n];
    h[i] = y > 0.0f ? y : 0.0f;
}

extern "C" void kernel_launch(void* const* d_in, const int* in_sizes, int n_in,
                              void* d_out, int out_size, void* d_ws, size_t ws_size,
                              hipStream_t stream) {
    const float* x    = (const float*)d_in[0];
    const int*   ei   = (const int*)d_in[1];
    const float* eps  = (const float*)d_in[2];
    const float* w0_1 = (const float*)d_in[3];
    const float* b0_1 = (const float*)d_in[4];
    const float* bn0g = (const float*)d_in[5];
    const float* bn0b = (const float*)d_in[6];
    const float* w0_2 = (const float*)d_in[7];
    const float* b0_2 = (const float*)d_in[8];
    const float* n0g  = (const float*)d_in[9];
    const float* n0b  = (const float*)d_in[10];
    const float* w1_1 = (const float*)d_in[11];
    const float* b1_1 = (const float*)d_in[12];
    const float* w1_2 = (const float*)d_in[13];
    const float* b1_2 = (const float*)d_in[14];
    const float* n1g  = (const float*)d_in[15];
    const float* n1b  = (const float*)d_in[16];
    const float* w2_1 = (const float*)d_in[17];
    const float* b2_1 = (const float*)d_in[18];
    const float* w2_2 = (const float*)d_in[19];
    const float* b2_2 = (const float*)d_in[20];
    const float* n2g  = (const float*)d_in[21];
    const float* n2b  = (const float*)d_in[22];

    const int N = in_sizes[0] / DHID;
    const int E = in_sizes[1] / 2;
    const int* src = ei;
    const int* dst = ei + E;

    float* ws0  = (float*)d_ws;                  // agg output
    float* ws1  = ws0 + (size_t)N * DHID;        // after first linear
    float* ws2  = ws1 + (size_t)N * DHID;        // after second linear (layer output)
    float* sums = ws2 + (size_t)N * DHID;        // 256 floats: colsum + colsumsq

    const int total = N * DHID;
    const int gEl   = (total + 255) / 256;
    const int gT    = (N + 16 * TILES_PER_BLOCK - 1) / (16 * TILES_PER_BLOCK);
    const int gE    = (E + 7) / 8;

    float* hout = (float*)d_out;

    // ---------------- layer 0 ----------------
    gin_scale<<<gEl, 256, 0, stream>>>(x, ws0, eps, total);
    gin_scatter<<<gE, 256, 0, stream>>>(x, src, dst, ws0, E);
    gin_zero<<<1, 256, 0, stream>>>(sums, 2 * DHID);
    gin_gemm<<<gT, 256, 0, stream>>>(ws0, w0_1, b0_1, ws1, sums, /*relu=*/0, N);
    gin_bn<<<gEl, 256, 0, stream>>>(ws1, sums, bn0g, bn0b, N);
    gin_zero<<<1, 256, 0, stream>>>(sums, 2 * DHID);
    gin_gemm<<<gT, 256, 0, stream>>>(ws1, w0_2, b0_2, ws2, sums, /*relu=*/0, N);
    gin_bn<<<gEl, 256, 0, stream>>>(ws2, sums, n0g, n0b, N);

    // ---------------- layer 1 ----------------
    gin_scale<<<gEl, 256, 0, stream>>>(ws2, ws0, eps, total);
    gin_scatter<<<gE, 256, 0, stream>>>(ws2, src, dst, ws0, E);
    gin_gemm<<<gT, 256, 0, stream>>>(ws0, w1_1, b1_1, ws1, nullptr, /*relu=*/1, N);
    gin_zero<<<1, 256, 0, stream>>>(sums, 2 * DHID);
    gin_gemm<<<gT, 256, 0, stream>>>(ws1, w1_2, b1_2, ws2, sums, /*relu=*/0, N);
    gin_bn<<<gEl, 256, 0, stream>>>(ws2, sums, n1g, n1b, N);

    // ---------------- layer 2 ----------------
    gin_scale<<<gEl, 256, 0, stream>>>(ws2, ws0, eps, total);
    gin_scatter<<<gE, 256, 0, stream>>>(ws2, src, dst, ws0, E);
    gin_gemm<<<gT, 256, 0, stream>>>(ws0, w2_1, b2_1, ws1, nullptr, /*relu=*/1, N);
    gin_zero<<<1, 256, 0, stream>>>(sums, 2 * DHID);
    gin_gemm<<<gT, 256, 0, stream>>>(ws1, w2_2, b2_2, hout, sums, /*relu=*/0, N);
    gin_bn<<<gEl, 256, 0, stream>>>(hout, sums, n2g, n2b, N);
}